// OctreeDeconvBnRelu_24988119728284
// MI455X (gfx1250) — compile-verified
//
#include <hip/hip_runtime.h>
#include <hip/hip_bf16.h>

#define N_NODES 131072
#define K_TAPS  27
#define CIN     32
#define COUT    32
#define BN_EPS  1e-5f

typedef float v2f __attribute__((ext_vector_type(2)));
typedef float v8f __attribute__((ext_vector_type(8)));

// LDS weight layout: [k][cout][cin] with cin-stride padded to 34 floats
// -> per-lane ds_load_b64 (cin pairs) is 8B aligned and bank-conflict free.
#define WSTRIDE 34
#define LDSW_PER_TAP (COUT * WSTRIDE)                 // 1088 floats
#define LDSW_FLOATS  (K_TAPS * LDSW_PER_TAP)          // 29376 floats = 117504 B

// ---------------------------------------------------------------------------
// K0: zero output accumulator + stats
// ---------------------------------------------------------------------------
__global__ void octdeconv_zero_kernel(float* __restrict__ out,
                                      float* __restrict__ stats, int n) {
  int i = blockIdx.x * blockDim.x + threadIdx.x;
  if (i < n) out[i] = 0.0f;
  if (i < 2 * COUT) stats[i] = 0.0f;
}

// ---------------------------------------------------------------------------
// K1: WMMA deconv + atomic scatter.
// One wave (32 lanes) owns a 16-row block of `data`. 8 waves / workgroup.
// ---------------------------------------------------------------------------
__global__ void octdeconv_wmma_kernel(const float* __restrict__ data,
                                      const float* __restrict__ weight,
                                      const int* __restrict__ neigh,
                                      float* __restrict__ out) {
  extern __shared__ float ldsW[];   // LDSW_FLOATS

  const int tid = threadIdx.x;

  // Cooperative load + transpose of all tap weights into LDS:
  // weight flat index = k*1024 + ci*32 + co  ->  ldsW[k][co][ci]
  for (int idx = tid; idx < K_TAPS * CIN * COUT; idx += blockDim.x) {
    int k  = idx >> 10;
    int r  = idx & 1023;
    int ci = r >> 5;
    int co = r & 31;
    ldsW[k * LDSW_PER_TAP + co * WSTRIDE + ci] = weight[idx];
  }
  __syncthreads();

  const int wave   = tid >> 5;
  const int lane   = tid & 31;
  const int laneLo = lane & 15;
  const int hi     = lane >> 4;                    // 0: K,K+1  1: K+2,K+3
  const int m0     = (blockIdx.x * 8 + wave) * 16; // first data row of tile
  const int row    = m0 + laneLo;

  // A fragments (16x4 f32, ISA layout): lane<16 -> v0=A[M][4s], v1=A[M][4s+1];
  // lane>=16 -> v0=A[M][4s+2], v1=A[M][4s+3]. Loaded once, reused 54x.
  v2f a[8];
  const float* drow = data + row * CIN + hi * 2;
#pragma unroll
  for (int s = 0; s < 8; ++s)
    a[s] = *(const v2f*)(drow + s * 4);

  for (int k = 0; k < K_TAPS; ++k) {
    // neigh[m0+laneLo][k]; per-row values broadcast later via shfl
    const int nb = neigh[(m0 + laneLo) * K_TAPS + k];

    const float* wk = ldsW + k * LDSW_PER_TAP;
    v8f acc0 = {0.f, 0.f, 0.f, 0.f, 0.f, 0.f, 0.f, 0.f};
    v8f acc1 = {0.f, 0.f, 0.f, 0.f, 0.f, 0.f, 0.f, 0.f};

#pragma unroll
    for (int s = 0; s < 8; ++s) {
      const int r0 = s * 4 + hi * 2;   // B rows this lane holds (K-dim)
      // B fragment (4x16 f32): lane L holds W[r0][col], W[r0+1][col]
      v2f b0 = *(const v2f*)(wk + (laneLo)      * WSTRIDE + r0); // cols 0..15
      v2f b1 = *(const v2f*)(wk + (laneLo + 16) * WSTRIDE + r0); // cols 16..31
      acc0 = __builtin_amdgcn_wmma_f32_16x16x4_f32(
          false, a[s], false, b0, (short)0, acc0, false, false);
      acc1 = __builtin_amdgcn_wmma_f32_16x16x4_f32(
          false, a[s], false, b1, (short)0, acc1, false, false);
    }

    // C/D layout: VGPR r, lanes 0-15 -> (row r, col lane); lanes 16-31 ->
    // (row r+8, col lane-16). Scatter-add each row to out[neigh[src][k]].
#pragma unroll
    for (int r = 0; r < 8; ++r) {
      const int srcRow = hi ? (r + 8) : r;
      const int dest   = __shfl(nb, srcRow, 32);
      atomicAdd(out + dest * COUT + laneLo,      acc0[r]);
      atomicAdd(out + dest * COUT + laneLo + 16, acc1[r]);
    }
  }
}

// ---------------------------------------------------------------------------
// K2: per-channel sum / sumsq reduction
// ---------------------------------------------------------------------------
__global__ void octdeconv_stats_kernel(const float* __restrict__ out,
                                       float* __restrict__ stats) {
  const int c  = threadIdx.x & 31;   // channel
  const int rg = threadIdx.x >> 5;   // row group 0..7
  float s = 0.f, s2 = 0.f;
  for (int row = blockIdx.x * 8 + rg; row < N_NODES; row += gridDim.x * 8) {
    float v = out[row * COUT + c];
    s  += v;
    s2 += v * v;
  }
  __shared__ float red[2][8][32];
  red[0][rg][c] = s;
  red[1][rg][c] = s2;
  __syncthreads();
  if (rg == 0) {
    float ts = 0.f, ts2 = 0.f;
#pragma unroll
    for (int j = 0; j < 8; ++j) { ts += red[0][j][c]; ts2 += red[1][j][c]; }
    atomicAdd(stats + c, ts);
    atomicAdd(stats + COUT + c, ts2);
  }
}

// ---------------------------------------------------------------------------
// K3: BatchNorm (batch stats) + ReLU, in place
// ---------------------------------------------------------------------------
__global__ void octdeconv_bnrelu_kernel(float* __restrict__ out,
                                        const float* __restrict__ stats,
                                        const float* __restrict__ gamma,
                                        const float* __restrict__ beta) {
  const int i = blockIdx.x * blockDim.x + threadIdx.x;
  const int c = i & 31;
  const float invN  = 1.0f / (float)N_NODES;
  const float mean  = stats[c] * invN;
  const float var   = stats[COUT + c] * invN - mean * mean;
  const float scale = gamma[c] * rsqrtf(var + BN_EPS);
  const float v     = (out[i] - mean) * scale + beta[c];
  out[i] = fmaxf(v, 0.0f);
}

// ---------------------------------------------------------------------------
extern "C" void kernel_launch(void* const* d_in, const int* in_sizes, int n_in,
                              void* d_out, int out_size, void* d_ws, size_t ws_size,
                              hipStream_t stream) {
  const float* data   = (const float*)d_in[0];
  const float* weight = (const float*)d_in[1];
  const float* gamma  = (const float*)d_in[2];
  const float* beta   = (const float*)d_in[3];
  const int*   neigh  = (const int*)d_in[4];
  float* out   = (float*)d_out;
  float* stats = (float*)d_ws;            // 64 floats

  const int total = N_NODES * COUT;       // 4,194,304

  octdeconv_zero_kernel<<<total / 256, 256, 0, stream>>>(out, stats, total);

  const int grid1 = N_NODES / (16 * 8);   // 1024 workgroups, 8 waves each
  octdeconv_wmma_kernel<<<grid1, 256, LDSW_FLOATS * sizeof(float), stream>>>(
      data, weight, neigh, out);

  octdeconv_stats_kernel<<<1024, 256, 0, stream>>>(out, stats);

  octdeconv_bnrelu_kernel<<<total / 256, 256, 0, stream>>>(out, stats, gamma, beta);
}